// SimpleMPNN_11175504904834
// MI455X (gfx1250) — compile-verified
//
#include <hip/hip_runtime.h>
#include <math.h>

typedef __attribute__((ext_vector_type(2))) float v2f;
typedef __attribute__((ext_vector_type(8))) float v8f;

// ---------------------------------------------------------------------------
// Kernel 0: qnorm = max(||q||, 1e-12)
// ---------------------------------------------------------------------------
__global__ void qnorm_kernel(const float* __restrict__ q, float* __restrict__ qn, int D) {
    __shared__ float red[256];
    float s = 0.0f;
    for (int i = threadIdx.x; i < D; i += blockDim.x) {
        float v = q[i];
        s += v * v;
    }
    red[threadIdx.x] = s;
    __syncthreads();
    for (int off = blockDim.x >> 1; off > 0; off >>= 1) {
        if ((int)threadIdx.x < off) red[threadIdx.x] += red[threadIdx.x + off];
        __syncthreads();
    }
    if (threadIdx.x == 0) qn[0] = fmaxf(sqrtf(red[0]), 1e-12f);
}

// ---------------------------------------------------------------------------
// Kernel 1: per-edge cosine similarity via V_WMMA_F32_16X16X4_F32.
// One wave32 handles 16 edges. A = (ea .* q) or (ea .* ea) in K=4 chunks,
// B = all-ones  =>  D[m][*] = rowsum(A_m), accumulated over D/4 chunks.
// A 16x4 f32 layout: lane m = lane&15, k = (lane>>4)*2 + vgpr   [ISA 7.12.2]
// C/D layout: VGPR r = row r (lanes 0-15) / row r+8 (lanes 16-31).
// ---------------------------------------------------------------------------
__global__ void edge_sim_kernel(const float* __restrict__ ea,
                                const float* __restrict__ q,
                                const float* __restrict__ qn,
                                float* __restrict__ sim,
                                int E, int D) {
    const int lane = threadIdx.x & 31;
    const int waveInBlock = threadIdx.x >> 5;
    const int tile = blockIdx.x * (blockDim.x >> 5) + waveInBlock;
    const int base = tile * 16;
    if (base >= E) return;  // wave-uniform exit

    int row = base + (lane & 15);
    if (row >= E) row = E - 1;               // clamp (keeps EXEC all-ones)
    const int koff = (lane >> 4) * 2;        // lanes 16-31 hold K=2,3 of each chunk

    const float* rp = ea + (long long)row * D + koff;
    const float* qp = q + koff;

    v8f c_eq = {};   // accumulates dot(ea, q) per row
    v8f c_ee = {};   // accumulates dot(ea, ea) per row
    v2f ones;
    ones.x = 1.0f;
    ones.y = 1.0f;

    for (int k0 = 0; k0 < D; k0 += 4) {
        v2f v  = *(const v2f*)(rp + k0);     // 8B aligned: (row*D + k0 + koff) even
        v2f qv = *(const v2f*)(qp + k0);
        v2f a_eq, a_ee;
        a_eq.x = v.x * qv.x;  a_eq.y = v.y * qv.y;
        a_ee.x = v.x * v.x;   a_ee.y = v.y * v.y;
        c_eq = __builtin_amdgcn_wmma_f32_16x16x4_f32(false, a_eq, false, ones,
                                                     (short)0, c_eq, false, false);
        c_ee = __builtin_amdgcn_wmma_f32_16x16x4_f32(false, a_ee, false, ones,
                                                     (short)0, c_ee, false, false);
    }

    const float qnv = qn[0];
    if (lane == 0 || lane == 16) {
        const int r0 = (lane == 0) ? 0 : 8;  // lane 0: rows 0-7, lane 16: rows 8-15
#pragma unroll
        for (int r = 0; r < 8; ++r) {
            const int m = base + r0 + r;
            if (m < E) {
                const float denom = fmaxf(sqrtf(c_ee[r]), 1e-12f) * qnv;
                sim[m] = c_eq[r] / denom;
            }
        }
    }
}

// ---------------------------------------------------------------------------
// Kernel 2: zero int fill
// ---------------------------------------------------------------------------
__global__ void zero_int_kernel(int* __restrict__ p, int n) {
    int i = blockIdx.x * blockDim.x + threadIdx.x;
    if (i < n) p[i] = 0;
}

// ---------------------------------------------------------------------------
// Kernel 3: in-degree counts (int atomics, only E of them)
// ---------------------------------------------------------------------------
__global__ void count_kernel(const long long* __restrict__ dstIdx,
                             int* __restrict__ cnt, int E) {
    int i = blockIdx.x * blockDim.x + threadIdx.x;
    if (i < E) atomicAdd(&cnt[(int)dstIdx[i]], 1);
}

// ---------------------------------------------------------------------------
// Kernel 4: single-block exclusive scan  cnt[N] -> rowptr[N+1], cursor[N]
// Hillis-Steele over 1024-wide chunks with a running carry.
// ---------------------------------------------------------------------------
__global__ void scan_kernel(const int* __restrict__ cnt,
                            int* __restrict__ rowptr,
                            int* __restrict__ cursor, int N) {
    __shared__ int sdata[1024];
    __shared__ int running;
    if (threadIdx.x == 0) running = 0;
    __syncthreads();
    for (int base = 0; base < N; base += blockDim.x) {
        const int i = base + threadIdx.x;
        const int v = (i < N) ? cnt[i] : 0;
        sdata[threadIdx.x] = v;
        __syncthreads();
        for (int off = 1; off < (int)blockDim.x; off <<= 1) {
            const int t = ((int)threadIdx.x >= off) ? sdata[threadIdx.x - off] : 0;
            __syncthreads();
            sdata[threadIdx.x] += t;
            __syncthreads();
        }
        const int incl = sdata[threadIdx.x];
        const int chunkTotal = sdata[blockDim.x - 1];
        const int carry = running;  // stable: last write was before previous barrier
        if (i < N) {
            const int pos = carry + incl - v;  // exclusive
            rowptr[i] = pos;
            cursor[i] = pos;
        }
        __syncthreads();
        if (threadIdx.x == 0) running = carry + chunkTotal;
        __syncthreads();
    }
    if (threadIdx.x == 0) rowptr[N] = running;
}

// ---------------------------------------------------------------------------
// Kernel 5: CSR placement — pack (src, sim) pairs in dst order.
// ---------------------------------------------------------------------------
__global__ void place_kernel(const long long* __restrict__ srcIdx,
                             const long long* __restrict__ dstIdx,
                             const float* __restrict__ sim,
                             int* __restrict__ cursor,
                             int* __restrict__ srcArr,
                             float* __restrict__ simArr, int E) {
    int e = blockIdx.x * blockDim.x + threadIdx.x;
    if (e >= E) return;
    const int d = (int)dstIdx[e];
    const int pos = atomicAdd(&cursor[d], 1);
    srcArr[pos] = (int)srcIdx[e];
    simArr[pos] = sim[e];
}

// ---------------------------------------------------------------------------
// Kernel 6: fused gather + mean + alpha-blend (atomic-free layer).
// 96 threads per node (D/4), each owns one float4 feature chunk.
// 96 = 3 waves => every wave is node-uniform: uniform loop bounds and
// broadcast (src, sim) loads; x gathers hit L2 (x is 15.4 MB << 192 MB).
// ---------------------------------------------------------------------------
__global__ void gather_combine_kernel(const float* __restrict__ xin,
                                      const int* __restrict__ rowptr,
                                      const int* __restrict__ srcArr,
                                      const float* __restrict__ simArr,
                                      float* __restrict__ xout,
                                      int N, int D, float alpha) {
    const int dq = D >> 2;
    long long tid = (long long)blockIdx.x * blockDim.x + threadIdx.x;
    if (tid >= (long long)N * dq) return;
    const int n  = (int)(tid / dq);
    const int fc = (int)(tid % dq) << 2;

    const int j0 = rowptr[n];
    const int j1 = rowptr[n + 1];

    float4 acc = make_float4(0.f, 0.f, 0.f, 0.f);
    for (int j = j0; j < j1; ++j) {
        const int s   = srcArr[j];
        const float w = simArr[j];
        const float4 v = *(const float4*)(xin + (long long)s * D + fc);
        acc.x += v.x * w;
        acc.y += v.y * w;
        acc.z += v.z * w;
        acc.w += v.w * w;
    }
    const float dinv = 1.0f / fmaxf((float)(j1 - j0), 1.0f);
    const float beta = 1.0f - alpha;
    const float4 xo = *(const float4*)(xin + tid * 4);
    float4 out;
    out.x = alpha * xo.x + beta * acc.x * dinv;
    out.y = alpha * xo.y + beta * acc.y * dinv;
    out.z = alpha * xo.z + beta * acc.z * dinv;
    out.w = alpha * xo.w + beta * acc.w * dinv;
    *(float4*)(xout + tid * 4) = out;
}

// ---------------------------------------------------------------------------
extern "C" void kernel_launch(void* const* d_in, const int* in_sizes, int n_in,
                              void* d_out, int out_size, void* d_ws, size_t ws_size,
                              hipStream_t stream) {
    const float*      x    = (const float*)d_in[0];       // [N, D] fp32
    const long long*  eidx = (const long long*)d_in[1];   // [2, E] int64
    const float*      ea   = (const float*)d_in[2];       // [E, D] fp32
    const float*      q    = (const float*)d_in[3];       // [D]    fp32

    const int D = in_sizes[3];
    const int N = in_sizes[0] / D;
    const int E = in_sizes[1] / 2;
    const long long* src = eidx;
    const long long* dst = eidx + E;

    // ---- workspace layout (256B aligned slices) ----
    char* ws = (char*)d_ws;
    size_t off = 0;
    auto align256 = [](size_t v) { return (v + 255) & ~(size_t)255; };
    float* qn     = (float*)(ws + off); off += 256;
    float* sim    = (float*)(ws + off); off += align256((size_t)E * 4);
    int*   cnt    = (int*)  (ws + off); off += align256((size_t)N * 4);
    int*   rowptr = (int*)  (ws + off); off += align256((size_t)(N + 1) * 4);
    int*   cursor = (int*)  (ws + off); off += align256((size_t)N * 4);
    int*   srcArr = (int*)  (ws + off); off += align256((size_t)E * 4);
    float* simArr = (float*)(ws + off); off += align256((size_t)E * 4);
    float* xtmp   = (float*)(ws + off); off += align256((size_t)N * D * 4);
    (void)ws_size; (void)n_in; (void)out_size;

    const int TB = 256;
    const long long gTot = (long long)N * (D >> 2);  // gather threads: N * 96

    // 0) ||q||
    qnorm_kernel<<<1, TB, 0, stream>>>(q, qn, D);

    // 1) per-edge cosine similarity (WMMA, 16 edges/wave, 8 waves/block)
    const int tiles = (E + 15) / 16;
    edge_sim_kernel<<<(tiles + 7) / 8, TB, 0, stream>>>(ea, q, qn, sim, E, D);

    // 2) CSR-by-dst build
    zero_int_kernel<<<(N + TB - 1) / TB, TB, 0, stream>>>(cnt, N);
    count_kernel<<<(E + TB - 1) / TB, TB, 0, stream>>>(dst, cnt, E);
    scan_kernel<<<1, 1024, 0, stream>>>(cnt, rowptr, cursor, N);
    place_kernel<<<(E + TB - 1) / TB, TB, 0, stream>>>(src, dst, sim, cursor,
                                                       srcArr, simArr, E);

    // 3) two atomic-free gather layers
    const int gBlocks = (int)((gTot + 191) / 192);  // 192 threads = 2 nodes/block
    gather_combine_kernel<<<gBlocks, 192, 0, stream>>>(x, rowptr, srcArr, simArr,
                                                       xtmp, N, D, 0.5f);
    gather_combine_kernel<<<gBlocks, 192, 0, stream>>>(xtmp, rowptr, srcArr, simArr,
                                                       (float*)d_out, N, D, 0.5f);
}